// UniPhyBlock_3788161155447
// MI455X (gfx1250) — compile-verified
//
#include <hip/hip_runtime.h>
#include <math.h>

// ---------------------------------------------------------------------------
// Problem constants
// ---------------------------------------------------------------------------
#define NB   2
#define NT   16
#define ND   64
#define NH   64
#define NW   64
#define NHW  4096          // H*W
#define NPOS 131072        // B*T*H*W
#define NCH  128           // 2*D (re|im concat)

// Workspace layout (float offsets). Total ~404 MB.
#define OFF_XN  0L                      // 131072*128  LN output / later reused as fr
#define OFF_XC  16777216L               // 131072*128  conv+residual (pos-major)
#define OFF_XE  33554432L               // 131072*128  encoded
#define OFF_UO  50331648L               // 131072*128  state-scan output
#define OFF_FF  67108864L               // 131072*256  FFN hidden
#define OFF_WC  100663296L              // 9*128*128   conv weight, fragment-swizzled
#define OFF_EB  100810752L              // 128*128     encode block matrix (swizzled)
#define OFF_ED  100827136L              // 128*128     decode block matrix (swizzled)
#define OFF_XM  100843520L              // 32*128      spatial mean
#define OFF_P   100847616L              // B*T*D*2
#define OFF_Q   100851712L              // B*T*D*2
#define OFF_A   100855808L              // B*T*D*2
#define OFF_W1S 100859904L              // 128*256     ffn_w1 swizzled
#define OFF_W2S 100892672L              // 256*128     ffn_w2 swizzled

typedef float v2f __attribute__((ext_vector_type(2)));
typedef float v8f __attribute__((ext_vector_type(8)));

__device__ __forceinline__ v8f wmma_f32(v2f a, v2f b, v8f c) {
  // D = A(16x4) * B(4x16) + C  -- V_WMMA_F32_16X16X4_F32
  return __builtin_amdgcn_wmma_f32_16x16x4_f32(false, a, false, b, (short)0, c,
                                               false, false);
}

// CDNA5 async global->LDS copy (ASYNCcnt path), 16 bytes per lane.
__device__ __forceinline__ void async_load_b128(const float* g, float* l) {
  unsigned laddr = (unsigned)(unsigned long long)(uintptr_t)l; // LDS aperture: low 32b = LDS offset
  unsigned long long gaddr = (unsigned long long)(uintptr_t)g;
  asm volatile("global_load_async_to_lds_b128 %0, %1, off"
               : : "v"(laddr), "v"(gaddr) : "memory");
}
__device__ __forceinline__ void wait_async0() {
  asm volatile("s_wait_asynccnt 0x0" : : : "memory");
}

// Fragment-major swizzle index for a KxN B-matrix:
// lane loads float2 at frag (kb, half, n) covering K = kb*4 + 2*half + {0,1}
__device__ __forceinline__ int swz(int k, int n, int N) {
  return (((k >> 2) * 2 + ((k >> 1) & 1)) * N + n) * 2 + (k & 1);
}

// ---------------------------------------------------------------------------
// K0: build + swizzle all weight matrices
// ---------------------------------------------------------------------------
__global__ void prep_weights(const float* __restrict__ wr, const float* __restrict__ wi,
                             const float* __restrict__ Ere, const float* __restrict__ Eim,
                             const float* __restrict__ Edre, const float* __restrict__ Edim,
                             const float* __restrict__ w1, const float* __restrict__ w2,
                             float* __restrict__ Wc2, float* __restrict__ Eb,
                             float* __restrict__ Edb, float* __restrict__ W1s,
                             float* __restrict__ W2s) {
  int idx = blockIdx.x * 256 + threadIdx.x;
  if (idx < 147456) {                    // conv: tap-blocked, K=i(128), N=o(128)
    int o = idx & 127;
    int i = (idx >> 7) & 127;
    int tap = idx >> 14;
    int ky = tap / 3, kx = tap % 3;
    float v;
    if (o < 64) {
      if (i < 64) v =  wr[((o * 64 + i) * 3 + ky) * 3 + kx];
      else        v = -wi[((o * 64 + (i - 64)) * 3 + ky) * 3 + kx];
    } else {
      if (i < 64) v =  wi[(((o - 64) * 64 + i) * 3 + ky) * 3 + kx];
      else        v =  wr[(((o - 64) * 64 + (i - 64)) * 3 + ky) * 3 + kx];
    }
    Wc2[tap * 16384 + swz(i, o, 128)] = v;
    return;
  }
  int j = idx - 147456;
  if (j < 16384) {                       // encode/decode block matrices (128x128)
    int col = j & 127, row = j >> 7;
    float v1 = (row < 64)
                 ? ((col < 64) ? Ere[row * 64 + col] : Eim[row * 64 + (col - 64)])
                 : ((col < 64) ? -Eim[(row - 64) * 64 + col]
                               : Ere[(row - 64) * 64 + (col - 64)]);
    float v2 = (row < 64)
                 ? ((col < 64) ? Edre[row * 64 + col] : Edim[row * 64 + (col - 64)])
                 : ((col < 64) ? -Edim[(row - 64) * 64 + col]
                               : Edre[(row - 64) * 64 + (col - 64)]);
    int f = swz(row, col, 128);
    Eb[f] = v1;
    Edb[f] = v2;
    return;
  }
  j -= 16384;
  if (j < 32768) {                       // ffn_w1: K=128, N=256
    int col = j & 255, row = j >> 8;
    W1s[swz(row, col, 256)] = w1[row * 256 + col];
    return;
  }
  j -= 32768;
  if (j < 32768) {                       // ffn_w2: K=256, N=128
    int col = j & 127, row = j >> 7;
    W2s[swz(row, col, 128)] = w2[row * 128 + col];
  }
}

// ---------------------------------------------------------------------------
// K1: LayerNorm over 128 channels + metric. One block per (bt,h) row; planar
// reads coalesced along w, transpose via LDS, wave computes 8 positions.
// ---------------------------------------------------------------------------
__global__ void ln_metric_kernel(const float* __restrict__ xre,
                                 const float* __restrict__ xim,
                                 const float* __restrict__ g,
                                 const float* __restrict__ b,
                                 const float* __restrict__ metric,
                                 float* __restrict__ xn) {
  __shared__ float Ls[64 * 130];
  int blk = blockIdx.x;                  // bt*64 + h
  int bt = blk >> 6, h = blk & 63;
  int tid = threadIdx.x;
  for (int idx = tid; idx < 128 * 64; idx += 256) {
    int c = idx >> 6, w = idx & 63;
    const float* src = (c < 64) ? xre : xim;
    Ls[w * 130 + c] = src[(bt * 64 + (c & 63)) * NHW + h * 64 + w];
  }
  __syncthreads();
  int wave = tid >> 5, lane = tid & 31;
  for (int pw = 0; pw < 8; ++pw) {
    int w = wave * 8 + pw;
    float v[4];
    float s = 0.f, sq = 0.f;
    for (int j = 0; j < 4; ++j) {
      int c = lane + 32 * j;
      float x = Ls[w * 130 + c];
      v[j] = x; s += x; sq += x * x;
    }
    for (int m = 16; m >= 1; m >>= 1) {
      s  += __shfl_xor(s, m, 32);
      sq += __shfl_xor(sq, m, 32);
    }
    float mean = s * (1.f / 128.f);
    float var  = sq * (1.f / 128.f) - mean * mean;
    float rstd = rsqrtf(var + 1e-5f);
    float mt = metric[h * 64 + w];
    int pos = bt * NHW + h * 64 + w;
    for (int j = 0; j < 4; ++j) {
      int c = lane + 32 * j;
      xn[pos * NCH + c] = ((v[j] - mean) * rstd * g[c] + b[c]) * mt;
    }
  }
}

// ---------------------------------------------------------------------------
// K2: 3x3 block-complex conv. Wave owns one 16-ch N column, 4 M accumulators
// so each B fragment (global b64) feeds 4 WMMAs; A fragments are ds b64.
// ---------------------------------------------------------------------------
__global__ void conv_kernel(const float* __restrict__ xn,
                            const float* __restrict__ Wc2,
                            const float* __restrict__ convb,
                            const float* __restrict__ xre,
                            const float* __restrict__ xim,
                            float* __restrict__ xc) {
  __shared__ __align__(16) float Xs[3 * 66 * 64];   // 49.5 KB
  int blk = blockIdx.x;                  // bt*64 + h
  int bt = blk >> 6;
  int h  = blk & 63;
  int tid = threadIdx.x;
  int wave = tid >> 5, lane = tid & 31;
  int half = lane >> 4, l15 = lane & 15;
  int n = wave * 16 + l15;               // output channel

  v8f acc[4];
  for (int mt = 0; mt < 4; ++mt) acc[mt] = (v8f){0,0,0,0,0,0,0,0};

  for (int cc = 0; cc < 2; ++cc) {       // input-channel chunk of 64
    __syncthreads();
    for (int idx = tid; idx < 3 * 66 * 16; idx += 256) {
      int c4 = idx & 15;
      int wp = (idx >> 4) % 66;
      int r  = idx / (66 * 16);
      int hh = h + r - 1;
      int w  = wp - 1;
      float4 val = {0.f, 0.f, 0.f, 0.f};
      if (hh >= 0 && hh < 64 && w >= 0 && w < 64)
        val = *(const float4*)&xn[(bt * NHW + hh * 64 + w) * NCH + cc * 64 + c4 * 4];
      *(float4*)&Xs[(r * 66 + wp) * 64 + c4 * 4] = val;
    }
    __syncthreads();
    for (int ky = 0; ky < 3; ++ky)
      for (int kx = 0; kx < 3; ++kx) {
        const v2f* Wt = (const v2f*)(Wc2 + (ky * 3 + kx) * 16384);
        for (int kb = 0; kb < 16; ++kb) {
          int kbg = cc * 16 + kb;
          v2f bfrag = Wt[(kbg * 2 + half) * 128 + n];
          for (int mt = 0; mt < 4; ++mt) {
            int wp = mt * 16 + l15 + kx;
            v2f a = *(const v2f*)&Xs[(ky * 66 + wp) * 64 + kb * 4 + 2 * half];
            acc[mt] = wmma_f32(a, bfrag, acc[mt]);
          }
        }
      }
  }
  // Epilogue: bias + residual with original planar x, store pos-major.
  int d = n & 63;
  const float* res = (n < 64) ? xre : xim;
  float cb = convb[n];
  for (int mt = 0; mt < 4; ++mt)
    for (int v = 0; v < 8; ++v) {
      int m = v + 8 * half;
      int w = mt * 16 + m;
      int pos = bt * NHW + h * 64 + w;
      xc[pos * NCH + n] = acc[mt][v] + cb + res[(bt * 64 + d) * NHW + h * 64 + w];
    }
}

// ---------------------------------------------------------------------------
// Generic tiled WMMA GEMM: C[131072 x N] = A[131072 x K] @ B[K x N] (swizzled).
// A tile staged via async global->LDS b128. Wave owns N columns, 4 M tiles.
// EPI 0: plain store.  EPI 1: +bias, exact GELU.  EPI 2: +bias +fr +xc,
//        transpose via LDS, coalesced planar write to (2,B,T,D,H,W).
// ---------------------------------------------------------------------------
template <int N, int K, int EPI>
__global__ void gemm_kernel(const float* __restrict__ A,
                            const float* __restrict__ Bs,
                            const float* __restrict__ bias,
                            float* __restrict__ C,
                            const float* __restrict__ fr,
                            const float* __restrict__ xc,
                            float* __restrict__ out) {
  __shared__ __align__(16) float As[64 * K];
  long row0 = (long)blockIdx.x * 64;
  int tid = threadIdx.x;
  // Warm weights into cache.
  for (int off = tid * 32; off < K * N; off += 256 * 32)
    __builtin_prefetch(Bs + off, 0, 3);
  // Async-stage the A tile (contiguous 64*K floats).
  {
    const float* src = A + row0 * K;
    for (int i = tid; i < (64 * K) / 4; i += 256)
      async_load_b128(src + i * 4, &As[i * 4]);
    wait_async0();
  }
  __syncthreads();
  int wave = tid >> 5, lane = tid & 31;
  int half = lane >> 4, l15 = lane & 15;
  const v2f* Bv = (const v2f*)Bs;
  for (int nt = wave; nt < N / 16; nt += 8) {
    int n = nt * 16 + l15;
    v8f acc[4];
    for (int mt = 0; mt < 4; ++mt) acc[mt] = (v8f){0,0,0,0,0,0,0,0};
    for (int kb = 0; kb < K / 4; ++kb) {
      v2f bfrag = Bv[(kb * 2 + half) * N + n];
      for (int mt = 0; mt < 4; ++mt) {
        v2f a = *(const v2f*)&As[(mt * 16 + l15) * K + kb * 4 + 2 * half];
        acc[mt] = wmma_f32(a, bfrag, acc[mt]);
      }
    }
    if (EPI == 2) __syncthreads();       // N=128 -> loop runs exactly once/wave
    for (int mt = 0; mt < 4; ++mt)
      for (int v = 0; v < 8; ++v) {
        int m = v + 8 * half;
        long pos = row0 + mt * 16 + m;
        float val = acc[mt][v];
        if (EPI == 0) {
          C[pos * N + n] = val;
        } else if (EPI == 1) {
          val += bias[n];
          val = 0.5f * val * (1.f + erff(val * 0.70710678118654752f));
          C[pos * N + n] = val;
        } else {
          val += bias[n] + fr[pos * NCH + n] + xc[pos * NCH + n];
          As[(mt * 16 + m) * 128 + n] = val;   // reuse As as transpose buffer
        }
      }
  }
  if (EPI == 2) {
    __syncthreads();
    int bt = (int)(row0 >> 12);
    int h64 = (int)(row0 & 4095);
    for (int idx = tid; idx < 64 * 128; idx += 256) {
      int w = idx & 63;
      int ch = idx >> 6;
      long o = ((long)bt * 64 + (ch & 63)) * NHW + h64 + w;
      if (ch >= 64) o += 8388608L;       // imag plane
      out[o] = As[w * 128 + ch];
    }
  }
}

// ---------------------------------------------------------------------------
// K4: spatial mean of xe over H*W -> xm[bt][128]
// ---------------------------------------------------------------------------
__global__ void reduce_xm(const float* __restrict__ xe, float* __restrict__ xm) {
  int bt = blockIdx.x;
  int c = threadIdx.x;   // 128 threads
  const float* p = xe + (long)bt * NHW * NCH + c;
  float s = 0.f;
  for (int i = 0; i < NHW; ++i) s += p[(long)i * NCH];
  xm[bt * NCH + c] = s * (1.f / (float)NHW);
}

// ---------------------------------------------------------------------------
// K5: flux scan over T, emits per-(b,t,d) complex coefficients P,Q,A and
// flux_out. 1 block, 128 threads = (b,d).
// ---------------------------------------------------------------------------
__global__ void flux_scan(const float* __restrict__ xm, const float* __restrict__ dt,
                          const float* __restrict__ lamf_re, const float* __restrict__ lamf_im,
                          const float* __restrict__ wsrc_re, const float* __restrict__ wsrc_im,
                          const float* __restrict__ g_w, const float* __restrict__ g_b,
                          const float* __restrict__ lam_re, const float* __restrict__ lam_im,
                          float* __restrict__ P, float* __restrict__ Q,
                          float* __restrict__ Aarr, float* __restrict__ out_flux) {
  int t0 = threadIdx.x;
  int b = t0 >> 6, d = t0 & 63;
  float lfr = lamf_re[d];
  float sp = (lfr > 20.f) ? lfr : log1pf(expf(lfr));
  float lf_re = -sp, lf_im = lamf_im[d];
  float lm_re = 0.3f * tanhf(lam_re[d]), lm_im = lam_im[d];
  float denf = lf_re * lf_re + lf_im * lf_im;
  float denl = lm_re * lm_re + lm_im * lm_im;
  float fre = 0.f, fim = 0.f;
  for (int t = 0; t < 16; ++t) {
    float dtn = dt[b * 16 + t];                     // TAU = 1
    float er  = expf(lf_re * dtn);
    float Afr = er * cosf(lf_im * dtn), Afi = er * sinf(lf_im * dtn);
    float nr = Afr - 1.f, ni = Afi;
    float Ffr = (nr * lf_re + ni * lf_im) / denf;
    float Ffi = (ni * lf_re - nr * lf_im) / denf;
    float xr = xm[(b * 16 + t) * NCH + d], xi = xm[(b * 16 + t) * NCH + 64 + d];
    float Xr = xr * Ffr - xi * Ffi, Xi = xr * Ffi + xi * Ffr;
    float tr = Afr * fre - Afi * fim + Xr;
    float ti = Afr * fim + Afi * fre + Xi;
    fre = tr; fim = ti;
    float sr = fre * wsrc_re[d] - fim * wsrc_im[d];
    float si = fre * wsrc_im[d] + fim * wsrc_re[d];
    float g = 1.f / (1.f + expf(-(fre * g_w[d] + g_b[d])));
    float er2 = expf(lm_re * dtn);
    float odr = er2 * cosf(lm_im * dtn), odi = er2 * sinf(lm_im * dtn);
    float n2r = odr - 1.f, n2i = odi;
    float ofr = (n2r * lm_re + n2i * lm_im) / denl;
    float ofi = (n2i * lm_re - n2r * lm_im) / denl;
    int o = (b * 16 + t) * 64 + d;
    P[2 * o]     = g * ofr;
    P[2 * o + 1] = g * ofi;
    float c1 = 1.f - g;
    Q[2 * o]     = (sr * ofr - si * ofi) * c1;
    Q[2 * o + 1] = (sr * ofi + si * ofr) * c1;
    Aarr[2 * o]     = odr;
    Aarr[2 * o + 1] = odi;
  }
  out_flux[b * 64 + d] = fre;
  out_flux[128 + b * 64 + d] = fim;
}

// ---------------------------------------------------------------------------
// K6: temporal state scan per (b,hw,e): uo[t] = A*uo[t-1] + (xe*P + Q [+ h*A@t0])
// ---------------------------------------------------------------------------
__global__ void state_scan(const float* __restrict__ xe,
                           const float* __restrict__ hre, const float* __restrict__ him,
                           const float* __restrict__ P, const float* __restrict__ Q,
                           const float* __restrict__ Aarr,
                           float* __restrict__ uo, float* __restrict__ out_h) {
  int tid = blockIdx.x * 256 + threadIdx.x;       // 0..524287
  int e  = tid & 63;
  int hw = (tid >> 6) & 4095;
  int b  = tid >> 18;
  float ur = 0.f, ui = 0.f;
  for (int t = 0; t < 16; ++t) {
    long pos = ((long)(b * 16 + t) << 12) + hw;
    float xr = xe[pos * NCH + e], xi = xe[pos * NCH + 64 + e];
    int o = (b * 16 + t) * 64 + e;
    float pr = P[2 * o], pi = P[2 * o + 1];
    float ar = Aarr[2 * o], ai = Aarr[2 * o + 1];
    float fr = xr * pr - xi * pi + Q[2 * o];
    float fi = xr * pi + xi * pr + Q[2 * o + 1];
    if (t == 0) {
      long hidx = ((long)b * 4096 + hw) * 64 + e;
      float h_r = hre[hidx], h_i = him[hidx];
      fr += h_r * ar - h_i * ai;
      fi += h_r * ai + h_i * ar;
    }
    float nr = ar * ur - ai * ui + fr;
    float ni = ar * ui + ai * ur + fi;
    ur = nr; ui = ni;
    uo[pos * NCH + e]      = ur;
    uo[pos * NCH + 64 + e] = ui;
  }
  long hidx = ((long)b * 4096 + hw) * 64 + e;
  out_h[hidx] = ur;
  out_h[524288L + hidx] = ui;
}

// ---------------------------------------------------------------------------
extern "C" void kernel_launch(void* const* d_in, const int* in_sizes, int n_in,
                              void* d_out, int out_size, void* d_ws, size_t ws_size,
                              hipStream_t stream) {
  const float* x_re    = (const float*)d_in[0];
  const float* x_im    = (const float*)d_in[1];
  const float* h_re    = (const float*)d_in[2];
  const float* h_im    = (const float*)d_in[3];
  const float* dt      = (const float*)d_in[4];
  const float* ln_g    = (const float*)d_in[5];
  const float* ln_b    = (const float*)d_in[6];
  const float* conv_wr = (const float*)d_in[7];
  const float* conv_wi = (const float*)d_in[8];
  const float* conv_b  = (const float*)d_in[9];
  const float* metric  = (const float*)d_in[10];
  const float* E_re    = (const float*)d_in[11];
  const float* E_im    = (const float*)d_in[12];
  const float* Edec_re = (const float*)d_in[13];
  const float* Edec_im = (const float*)d_in[14];
  const float* lam_re  = (const float*)d_in[15];
  const float* lam_im  = (const float*)d_in[16];
  const float* lamf_re = (const float*)d_in[17];
  const float* lamf_im = (const float*)d_in[18];
  const float* wsrc_re = (const float*)d_in[19];
  const float* wsrc_im = (const float*)d_in[20];
  const float* g_w     = (const float*)d_in[21];
  const float* g_b     = (const float*)d_in[22];
  const float* ffn_w1  = (const float*)d_in[23];
  const float* ffn_b1  = (const float*)d_in[24];
  const float* ffn_w2  = (const float*)d_in[25];
  const float* ffn_b2  = (const float*)d_in[26];

  float* ws  = (float*)d_ws;
  float* out = (float*)d_out;

  prep_weights<<<896, 256, 0, stream>>>(conv_wr, conv_wi, E_re, E_im, Edec_re,
                                        Edec_im, ffn_w1, ffn_w2,
                                        ws + OFF_WC, ws + OFF_EB, ws + OFF_ED,
                                        ws + OFF_W1S, ws + OFF_W2S);
  ln_metric_kernel<<<NB * NT * NH, 256, 0, stream>>>(x_re, x_im, ln_g, ln_b,
                                                     metric, ws + OFF_XN);
  conv_kernel<<<NB * NT * NH, 256, 0, stream>>>(ws + OFF_XN, ws + OFF_WC, conv_b,
                                                x_re, x_im, ws + OFF_XC);
  // encode: xe = xc @ Eblk
  gemm_kernel<128, 128, 0><<<NPOS / 64, 256, 0, stream>>>(
      ws + OFF_XC, ws + OFF_EB, nullptr, ws + OFF_XE, nullptr, nullptr, nullptr);
  reduce_xm<<<NB * NT, 128, 0, stream>>>(ws + OFF_XE, ws + OFF_XM);
  flux_scan<<<1, 128, 0, stream>>>(ws + OFF_XM, dt, lamf_re, lamf_im, wsrc_re,
                                   wsrc_im, g_w, g_b, lam_re, lam_im,
                                   ws + OFF_P, ws + OFF_Q, ws + OFF_A,
                                   out + 17825792L);
  state_scan<<<2048, 256, 0, stream>>>(ws + OFF_XE, h_re, h_im, ws + OFF_P,
                                       ws + OFF_Q, ws + OFF_A, ws + OFF_UO,
                                       out + 16777216L);
  // decode: fr = uo @ Edecblk (reuse XN region as fr)
  gemm_kernel<128, 128, 0><<<NPOS / 64, 256, 0, stream>>>(
      ws + OFF_UO, ws + OFF_ED, nullptr, ws + OFF_XN, nullptr, nullptr, nullptr);
  // FFN1: ff = gelu(fr @ W1 + b1)
  gemm_kernel<256, 128, 1><<<NPOS / 64, 256, 0, stream>>>(
      ws + OFF_XN, ws + OFF_W1S, ffn_b1, ws + OFF_FF, nullptr, nullptr, nullptr);
  // FFN2 + fc = fr + ff2 ; y = xc + fc -> planar output
  gemm_kernel<128, 256, 2><<<NPOS / 64, 256, 0, stream>>>(
      ws + OFF_FF, ws + OFF_W2S, ffn_b2, nullptr, ws + OFF_XN, ws + OFF_XC, out);
}